// Net_77841987273494
// MI455X (gfx1250) — compile-verified
//
#include <hip/hip_runtime.h>
#include <hip/hip_bf16.h>

typedef float v2f __attribute__((ext_vector_type(2)));
typedef float v8f __attribute__((ext_vector_type(8)));

#define DIN   128
#define DMID  128
#define DATTR 32
#define DOUT  128
#define DEMB  256
#define DH    768
#define KSEL  16
#define DCAT  160   // DMID+DATTR == DOUT+DATTR

// ---------------------------------------------------------------- utilities
__global__ void k_fill(float* __restrict__ p, int n, float v) {
    int t = blockIdx.x * blockDim.x + threadIdx.x;
    if (t < n) p[t] = v;
}

// deg[dst] += w  (deg pre-filled with 1.0 for the self loop)
__global__ void k_deg(const long long* __restrict__ dst,
                      const float* __restrict__ w,
                      float* __restrict__ deg, int E) {
    int e = blockIdx.x * blockDim.x + threadIdx.x;
    if (e < E) atomicAdd(&deg[(int)dst[e]], w[e]);
}

__global__ void k_dinv(float* __restrict__ d, int n) {
    int t = blockIdx.x * blockDim.x + threadIdx.x;
    if (t < n) {
        float v = d[t];
        d[t] = (v > 0.0f) ? rsqrtf(fmaxf(v, 1e-12f)) : 0.0f;
    }
}

__global__ void k_norm(const long long* __restrict__ src,
                       const long long* __restrict__ dst,
                       const float* __restrict__ w,
                       const float* __restrict__ dinv,
                       float* __restrict__ norm, int E) {
    int e = blockIdx.x * blockDim.x + threadIdx.x;
    if (e < E) norm[e] = dinv[(int)src[e]] * w[e] * dinv[(int)dst[e]];
}

// ---------------------------------------------------------- WMMA fp32 GEMM
// C[M x 128] = A[M x Kdim] (row-major, stride lda) @ B[Kdim x 128] (row-major)
// One wave per 16x16 tile; V_WMMA_F32_16X16X4_F32, K loop of 4.
__global__ void k_gemm_wmma(const float* __restrict__ A,
                            const float* __restrict__ B,
                            float* __restrict__ C,
                            int M, int Kdim, int lda) {
    const int NT = DOUT / 16;                       // 8 n-tiles
    int wave = (blockIdx.x * blockDim.x + threadIdx.x) >> 5;
    int lane = threadIdx.x & 31;
    int mt = wave / NT;
    int nt = wave % NT;
    if (mt * 16 >= M) return;                       // uniform per wave

    int half = lane >> 4;                           // 0: K0/K1, 1: K2/K3
    int l16  = lane & 15;
    long long arow = (long long)(mt * 16 + l16);
    int bcol = nt * 16 + l16;

    v8f c = {0.f, 0.f, 0.f, 0.f, 0.f, 0.f, 0.f, 0.f};

    for (int k = 0; k < Kdim; k += 4) {
        int ka = k + half * 2;
        v2f a, b;
        a.x = A[arow * lda + ka + 0];
        a.y = A[arow * lda + ka + 1];
        b.x = B[(long long)(ka + 0) * DOUT + bcol];
        b.y = B[(long long)(ka + 1) * DOUT + bcol];
        c = __builtin_amdgcn_wmma_f32_16x16x4_f32(
                /*neg_a=*/false, a, /*neg_b=*/false, b,
                /*c_mod=*/(short)0, c, /*reuse_a=*/false, /*reuse_b=*/false);
    }

    // C/D layout: VGPR r -> M = r + 8*half, N = l16
    long long row0 = (long long)(mt * 16 + half * 8);
    int col = nt * 16 + l16;
#pragma unroll
    for (int r = 0; r < 8; ++r)
        C[(row0 + r) * DOUT + col] = c[r];
}

// ------------------------------------------------------------- aggregation
// out[i*160 + j] = bias[j] + tmp[i*128 + j] * dinv[i]^2   (self loop + bias)
__global__ void k_init_agg(const float* __restrict__ tmp,
                           const float* __restrict__ dinv,
                           const float* __restrict__ bias,
                           float* __restrict__ out, int N) {
    int t = blockIdx.x * blockDim.x + threadIdx.x;
    int i = t >> 7;                  // /128
    int j = t & 127;
    if (i >= N) return;
    float dv = dinv[i];
    out[(long long)i * DCAT + j] = bias[j] + tmp[(long long)i * DOUT + j] * dv * dv;
}

// 32 threads per edge, 4 floats each: out[dst] += tmp[src]*norm
__global__ void k_aggregate(const float* __restrict__ tmp,
                            const long long* __restrict__ src,
                            const long long* __restrict__ dst,
                            const float* __restrict__ norm,
                            float* __restrict__ out, int E) {
    int t = blockIdx.x * blockDim.x + threadIdx.x;
    int e = t >> 5;
    if (e >= E) return;
    int f = (t & 31) * 4;
    int s = (int)src[e], d = (int)dst[e];
    float nv = norm[e];
    const float* hs = tmp + (long long)s * DOUT + f;
    float* od = out + (long long)d * DCAT + f;
    atomicAdd(&od[0], hs[0] * nv);
    atomicAdd(&od[1], hs[1] * nv);
    atomicAdd(&od[2], hs[2] * nv);
    atomicAdd(&od[3], hs[3] * nv);
}

// optional relu on cols [0,128), copy attributes into cols [128,160)
__global__ void k_finalize(const float* __restrict__ attr,
                           float* __restrict__ out, int N, int do_relu) {
    int t = blockIdx.x * blockDim.x + threadIdx.x;
    int i = t / DCAT;
    int j = t % DCAT;
    if (i >= N) return;
    long long o = (long long)i * DCAT + j;
    if (j < DOUT) {
        if (do_relu) { float v = out[o]; out[o] = v > 0.0f ? v : 0.0f; }
    } else {
        out[o] = attr[(long long)i * DATTR + (j - DOUT)];
    }
}

// ------------------------------------------------------------ head kernels
__global__ void k_q(const float* __restrict__ hidden,
                    const float* __restrict__ Wattn,
                    float* __restrict__ q) {
    int j = blockIdx.x * blockDim.x + threadIdx.x;
    if (j >= DCAT) return;
    float s = 0.0f;
    for (int k = 0; k < DH; ++k) s += hidden[k] * Wattn[(long long)k * DCAT + j];
    q[j] = s;
}

__global__ void k_scores(const float* __restrict__ H,
                         const float* __restrict__ q,
                         float* __restrict__ scores, int N) {
    int wave = (blockIdx.x * blockDim.x + threadIdx.x) >> 5;
    int lane = threadIdx.x & 31;
    if (wave >= N) return;
    const float* h = H + (long long)wave * DCAT;
    float s = 0.0f;
    for (int j = lane; j < DCAT; j += 32) s += h[j] * q[j];
#pragma unroll
    for (int off = 16; off > 0; off >>= 1) s += __shfl_down(s, off, 32);
    if (lane == 0) scores[wave] = s;
}

// single-block iterative argmax top-K (ties -> smaller index, matching top_k)
__global__ void k_topk(float* __restrict__ scores, int* __restrict__ idx_out, int N) {
    __shared__ float sv[1024];
    __shared__ int   si[1024];
    int tid = threadIdx.x;
    for (int k = 0; k < KSEL; ++k) {
        float best = -3.402823e38f; int bi = 0x7fffffff;
        for (int i = tid; i < N; i += blockDim.x) {
            float v = scores[i];
            if (v > best) { best = v; bi = i; }
        }
        sv[tid] = best; si[tid] = bi;
        __syncthreads();
        for (int s = blockDim.x >> 1; s > 0; s >>= 1) {
            if (tid < s) {
                if (sv[tid + s] > sv[tid] ||
                    (sv[tid + s] == sv[tid] && si[tid + s] < si[tid])) {
                    sv[tid] = sv[tid + s];
                    si[tid] = si[tid + s];
                }
            }
            __syncthreads();
        }
        if (tid == 0) { idx_out[k] = si[0]; scores[si[0]] = -3.402823e38f; }
        __syncthreads();
    }
}

__global__ void k_gather(const float* __restrict__ H,
                         const int* __restrict__ idx,
                         float* __restrict__ out) {
    int t = blockIdx.x * blockDim.x + threadIdx.x;   // KSEL*DCAT = 2560
    if (t >= KSEL * DCAT) return;
    int k = t / DCAT, j = t % DCAT;
    out[DEMB + t] = H[(long long)idx[k] * DCAT + j];
}

__global__ void k_colsum(const float* __restrict__ H,
                         float* __restrict__ colsum, int N, int rowsPerBlock) {
    int j = threadIdx.x;                              // blockDim == DCAT
    long long r0 = (long long)blockIdx.x * rowsPerBlock;
    long long r1 = r0 + rowsPerBlock; if (r1 > N) r1 = N;
    float s = 0.0f;
    for (long long r = r0; r < r1; ++r) s += H[r * DCAT + j];
    atomicAdd(&colsum[j], s);
}

__global__ void k_emb(const float* __restrict__ colsum,
                      const float* __restrict__ Wmap,
                      const float* __restrict__ bmap,
                      float* __restrict__ out, int N) {
    int j = blockIdx.x * blockDim.x + threadIdx.x;
    if (j >= DEMB) return;
    float invN = 1.0f / (float)N;
    float s = bmap[j];
    for (int k = 0; k < DCAT; ++k) s += (colsum[k] * invN) * Wmap[(long long)k * DEMB + j];
    out[j] = s;
}

// ---------------------------------------------------------------- launcher
extern "C" void kernel_launch(void* const* d_in, const int* in_sizes, int n_in,
                              void* d_out, int out_size, void* d_ws, size_t ws_size,
                              hipStream_t stream) {
    const float*     x      = (const float*)d_in[0];
    const float*     attr   = (const float*)d_in[1];
    const float*     ew     = (const float*)d_in[2];
    const float*     hidden = (const float*)d_in[3];
    const long long* eidx   = (const long long*)d_in[4];
    const float*     W1     = (const float*)d_in[5];
    const float*     b1     = (const float*)d_in[6];
    const float*     W2     = (const float*)d_in[7];
    const float*     b2     = (const float*)d_in[8];
    const float*     Wmap   = (const float*)d_in[9];
    const float*     bmap   = (const float*)d_in[10];
    const float*     Wattn  = (const float*)d_in[11];
    float* out = (float*)d_out;

    const int N = in_sizes[0] / DIN;
    const int E = in_sizes[2];
    const long long* src = eidx;
    const long long* dst = eidx + E;

    // workspace carve-up (floats)
    float* ws     = (float*)d_ws;
    float* dinv   = ws;                       size_t o = (size_t)N;
    float* norm   = ws + o;                   o += (size_t)E;
    float* tmp    = ws + o;                   o += (size_t)N * DOUT;
    float* hcat1  = ws + o;                   o += (size_t)N * DCAT;
    float* Hm     = ws + o;                   o += (size_t)N * DCAT;
    float* scores = ws + o;                   o += (size_t)N;
    float* qv     = ws + o;                   o += (size_t)DCAT;
    float* colsum = ws + o;                   o += (size_t)DCAT;
    int*   topk   = (int*)(ws + o);

    const int B = 256;
    auto nb = [](long long n, int b) { return (int)((n + b - 1) / b); };

    // --- degree / norms (shared by both conv layers) ---
    k_fill<<<nb(N, B), B, 0, stream>>>(dinv, N, 1.0f);        // self-loop weight
    k_deg<<<nb(E, B), B, 0, stream>>>(dst, ew, dinv, E);
    k_dinv<<<nb(N, B), B, 0, stream>>>(dinv, N);
    k_norm<<<nb(E, B), B, 0, stream>>>(src, dst, ew, dinv, norm, E);

    // --- layer 1: tmp = x @ W1 (WMMA), aggregate into hcat1, relu+concat ---
    {
        long long waves = (long long)((N + 15) / 16) * (DOUT / 16);
        k_gemm_wmma<<<nb(waves * 32, B), B, 0, stream>>>(x, W1, tmp, N, DIN, DIN);
    }
    k_init_agg<<<nb((long long)N * DOUT, B), B, 0, stream>>>(tmp, dinv, b1, hcat1, N);
    k_aggregate<<<nb((long long)E * 32, B), B, 0, stream>>>(tmp, src, dst, norm, hcat1, E);
    k_finalize<<<nb((long long)N * DCAT, B), B, 0, stream>>>(attr, hcat1, N, 1);

    // --- layer 2: tmp = hcat1 @ W2 (WMMA), aggregate into H, concat ---
    {
        long long waves = (long long)((N + 15) / 16) * (DOUT / 16);
        k_gemm_wmma<<<nb(waves * 32, B), B, 0, stream>>>(hcat1, W2, tmp, N, DCAT, DCAT);
    }
    k_init_agg<<<nb((long long)N * DOUT, B), B, 0, stream>>>(tmp, dinv, b2, Hm, N);
    k_aggregate<<<nb((long long)E * 32, B), B, 0, stream>>>(tmp, src, dst, norm, Hm, E);
    k_finalize<<<nb((long long)N * DCAT, B), B, 0, stream>>>(attr, Hm, N, 0);

    // --- attention head: q, scores, top-16, gather ---
    k_q<<<1, DCAT, 0, stream>>>(hidden, Wattn, qv);
    k_scores<<<nb((long long)N * 32, B), B, 0, stream>>>(Hm, qv, scores, N);
    k_topk<<<1, 1024, 0, stream>>>(scores, topk, N);
    k_gather<<<nb(KSEL * DCAT, B), B, 0, stream>>>(Hm, topk, out);

    // --- mean-pool + final mapping ---
    k_fill<<<1, DCAT, 0, stream>>>(colsum, DCAT, 0.0f);
    {
        int rowsPerBlock = 256;
        k_colsum<<<nb(N, rowsPerBlock), DCAT, 0, stream>>>(Hm, colsum, N, rowsPerBlock);
    }
    k_emb<<<1, DEMB, 0, stream>>>(colsum, Wmap, bmap, out, N);
}